// GPDGaussian_70111046140123
// MI455X (gfx1250) — compile-verified
//
#include <hip/hip_runtime.h>
#include <hip/hip_bf16.h>
#include <math.h>

// ---------------------------------------------------------------------------
// GPD Gaussian head for MI455X (gfx1250, wave32).
//   m  = (W@x+b)[0:34]                         -> (B,H,W,34)
//   sp = smin*(1-sig)+smax*sig of next 34      -> (B,34,H,W)
//   a  = pi*tanh of last 561                   -> Givens angles
//   R  = prod of 561 Givens rotations (34x34, per pixel)  [register-resident]
//   S  = R^T diag(sp) R                        -> (B,H,W,34,34)  [WMMA fp32]
// One wave per pixel; 4 waves per block; fused single kernel.
// ---------------------------------------------------------------------------

typedef __attribute__((ext_vector_type(2))) float v2f;
typedef __attribute__((ext_vector_type(8))) float v8f;

#define C_CH 34
#define A_CH 561
#define NOUT 629            // C + C + A
#define RSTRIDE 36          // padded LDS row stride for R/Rs (bank-friendly)
#define PIX_PER_BLOCK 4
#define HW 4096             // 64*64
#define SMIN_V 0.001f
#define SMAX_V 1000.0f
#define PI_F 3.14159265358979323846f

__global__ __launch_bounds__(32 * PIX_PER_BLOCK)
void gpd_gaussian_kernel(const float* __restrict__ x,
                         const float* __restrict__ Wm,
                         const float* __restrict__ bv,
                         float* __restrict__ out_m,
                         float* __restrict__ out_S,
                         float* __restrict__ out_s)
{
    __shared__ float lds_R   [PIX_PER_BLOCK][C_CH * RSTRIDE]; // R
    __shared__ float lds_Rs  [PIX_PER_BLOCK][C_CH * RSTRIDE]; // diag(sp)*R
    __shared__ float lds_csss[PIX_PER_BLOCK][2 * A_CH];       // interleaved cos,sin
    __shared__ float lds_sp  [PIX_PER_BLOCK][C_CH];

    const int lane = threadIdx.x & 31;
    const int wv   = threadIdx.x >> 5;                 // wave slot = pixel slot
    const int pix  = blockIdx.x * PIX_PER_BLOCK + wv;  // 0..16383
    const int bIdx = pix >> 12;                        // pixel / 4096
    const int hw   = pix & (HW - 1);

    // --- 3 input channels for this pixel (uniform across the wave) ---------
    const float x0 = x[(bIdx * 3 + 0) * HW + hw];
    const float x1 = x[(bIdx * 3 + 1) * HW + hw];
    const float x2 = x[(bIdx * 3 + 2) * HW + hw];

    // --- encoder: 629 channels, lane-strided -------------------------------
    for (int o = lane; o < NOUT; o += 32) {
        float w = Wm[o * 3 + 0] * x0 + Wm[o * 3 + 1] * x1
                + Wm[o * 3 + 2] * x2 + bv[o];
        if (o < C_CH) {
            out_m[(size_t)pix * C_CH + o] = w;                      // mean
        } else if (o < 2 * C_CH) {
            int c = o - C_CH;
            float sg = 1.0f / (1.0f + __expf(-w));                  // sigmoid
            float sp = SMIN_V * (1.0f - sg) + SMAX_V * sg;
            out_s[((size_t)bIdx * C_CH + c) * HW + hw] = sp;
            lds_sp[wv][c] = sp;
        } else {
            int t = o - 2 * C_CH;
            float ang = PI_F * tanhf(w);
            lds_csss[wv][2 * t]     = __cosf(ang);
            lds_csss[wv][2 * t + 1] = __sinf(ang);
        }
    }
    __syncthreads();   // csss/sp visible to all lanes

    // --- Givens scan, fully unrolled with the row in registers -------------
    // Rows of R are independent under column rotations: lane owns rows
    // m = lane (+32).  After unrolling, r[i]/r[j] indices are constants, so
    // the whole 561-step chain is register FMAs + one ds_load_b64 per step.
    for (int m = lane; m < C_CH; m += 32) {
        float r[C_CH];
#pragma unroll
        for (int k = 0; k < C_CH; ++k) r[k] = (k == m) ? 1.0f : 0.0f;

        int t = 0;
#pragma unroll
        for (int i = 0; i < C_CH - 1; ++i) {
#pragma unroll
            for (int j = i + 1; j < C_CH; ++j, ++t) {
                float c  = lds_csss[wv][2 * t];
                float s  = lds_csss[wv][2 * t + 1];
                float ri = r[i], rj = r[j];
                r[j] = fmaf(c, rj, -s * ri);   // -s*ri + c*rj
                r[i] = fmaf(c, ri,  s * rj);   //  c*ri + s*rj
            }
        }

        const float spm = lds_sp[wv][m];
#pragma unroll
        for (int k = 0; k < C_CH; ++k) {
            lds_R [wv][m * RSTRIDE + k] = r[k];
            lds_Rs[wv][m * RSTRIDE + k] = spm * r[k];
        }
    }
    __syncthreads();   // full R / Rs visible to all lanes of the wave

    // --- S = R^T * Rs via V_WMMA_F32_16X16X4_F32 ---------------------------
    // Padded to 48x48 output, K padded to 36 (9 steps of 4).
    // Fragment layout (fp32, wave32):
    //   A 16x4 : vgpr0 -> K = 2*half,   vgpr1 -> K = 2*half+1,  M = lane&15
    //   B 4x16 : vgpr0 -> K = 2*half,   vgpr1 -> K = 2*half+1,  N = lane&15
    //   C/D    : vgpr v -> M = 16*mt + v + 8*half,              N = lane&15
    const int l  = lane & 15;
    const int hf = lane >> 4;

    for (int mt = 0; mt < 3; ++mt) {
        for (int nt = 0; nt < 3; ++nt) {
            v8f acc = {0.f, 0.f, 0.f, 0.f, 0.f, 0.f, 0.f, 0.f};
            const int  kcol = mt * 16 + l;           // S row  (col of R)
            const int  ncol = nt * 16 + l;           // S col  (col of R)
            const bool kok  = (kcol < C_CH);
            const bool nok  = (ncol < C_CH);
#pragma unroll
            for (int kk = 0; kk < 9; ++kk) {
                const int  m0   = kk * 4 + 2 * hf;   // K element pair
                const int  m1   = m0 + 1;
                const bool m0ok = (m0 < C_CH);
                const bool m1ok = (m1 < C_CH);
                v2f a, b;
                // A = R^T tile:        A[Mrow,Kc] = R [m][kcol]
                a.x = (kok && m0ok) ? lds_R [wv][m0 * RSTRIDE + kcol] : 0.0f;
                a.y = (kok && m1ok) ? lds_R [wv][m1 * RSTRIDE + kcol] : 0.0f;
                // B = diag(sp)*R tile: B[Kr,Nc]  = Rs[m][ncol]
                b.x = (nok && m0ok) ? lds_Rs[wv][m0 * RSTRIDE + ncol] : 0.0f;
                b.y = (nok && m1ok) ? lds_Rs[wv][m1 * RSTRIDE + ncol] : 0.0f;
                acc = __builtin_amdgcn_wmma_f32_16x16x4_f32(
                        /*neg_a=*/false, a, /*neg_b=*/false, b,
                        /*c_mod=*/(short)0, acc,
                        /*reuse_a=*/false, /*reuse_b=*/false);
            }
            // scatter the 16x16 tile into the 34x34 S block of this pixel
            float* Sp = out_S + (size_t)pix * (C_CH * C_CH);
#pragma unroll
            for (int v = 0; v < 8; ++v) {
                int row = mt * 16 + v + 8 * hf;
                if (row < C_CH && ncol < C_CH)
                    Sp[row * C_CH + ncol] = acc[v];
            }
        }
    }
}

extern "C" void kernel_launch(void* const* d_in, const int* in_sizes, int n_in,
                              void* d_out, int out_size, void* d_ws, size_t ws_size,
                              hipStream_t stream) {
    (void)in_sizes; (void)n_in; (void)d_ws; (void)ws_size; (void)out_size;
    const float* x  = (const float*)d_in[0];   // (4,3,64,64)
    const float* W  = (const float*)d_in[1];   // (629,3)
    const float* b  = (const float*)d_in[2];   // (629,)

    const size_t n_m = (size_t)4 * HW * C_CH;          //   557056
    const size_t n_S = (size_t)4 * HW * C_CH * C_CH;   // 18939904
    float* out_m = (float*)d_out;
    float* out_S = out_m + n_m;
    float* out_s = out_S + n_S;

    const int npix   = 4 * HW;                         // 16384
    const int blocks = npix / PIX_PER_BLOCK;           // 4096
    gpd_gaussian_kernel<<<blocks, 32 * PIX_PER_BLOCK, 0, stream>>>(
        x, W, b, out_m, out_S, out_s);
}